// SSIM_51101520888575
// MI455X (gfx1250) — compile-verified
//
#include <hip/hip_runtime.h>

// ---------------------------------------------------------------------------
// SSIM (skimage-style, win=7, valid) for 128 x (384x384) fp32 image pairs.
// Box filters computed as banded matmuls on the CDNA5 matrix pipe using
// V_WMMA_F32_16X16X4_F32 (vertical 7-tap, then horizontal 7-tap).
// Per-wave: one 16-wide block, all 5 statistics (x, y, x^2, y^2, xy) at once.
// ---------------------------------------------------------------------------

typedef float v2f __attribute__((ext_vector_type(2)));
typedef float v8f __attribute__((ext_vector_type(8)));

#define IMG        384
#define OUTD       378          // 384 - 6 (valid conv)
#define NIMG       128          // 32 * 4 planes
#define STRIPE_W   64           // output cols per column-stripe
#define NSTRIPE    6            // 6*64 = 384 >= 378
#define STRIP_H    126          // output rows per row-strip (3*126 == 378 exactly)
#define NSTRIP     3
#define NB         8            // row batches of 16 per strip (7*16 + 14)
#define LROWS      22           // input rows needed per 16-row batch
#define LROWS_PAD  24           // padded to cover K=24 (rows 22,23 zeroed)
#define INCOLS     70           // input cols needed per stripe (64 + 6)
#define LXW        80           // padded staging width
#define SCW        83           // Scol row stride (odd -> conflict-free A reads)
#define HW         67           // H row stride
#define SCOL_FLOATS (5 * 16 * SCW)        // 6640
#define UOFF        SCOL_FLOATS
#define LX_OFF      UOFF
#define LY_OFF      (UOFF + LROWS_PAD * LXW)   // staging: 2*24*80 = 3840 floats
#define H_OFF       UOFF                       // union: H overlays dead staging
#define HSTAT       (16 * HW)
#define UNION_FLOATS (5 * 16 * HW)             // 5360 > 3840
#define SMEM_FLOATS (SCOL_FLOATS + UNION_FLOATS)   // 12000 floats = 48 KB
#define NBLK        (NIMG * NSTRIPE * NSTRIP)      // 2304

// band weight: W[m][k] = 1 iff k-m in [0,6]  (vertical A == horizontal B)
__device__ __forceinline__ float band7(int m, int k) {
  int d = k - m;
  return (d >= 0 && d <= 6) ? 1.0f : 0.0f;
}

__device__ __forceinline__ v8f wmma4(v2f a, v2f b, v8f c) {
  // D(16x16,f32) = A(16x4,f32) * B(4x16,f32) + C
  return __builtin_amdgcn_wmma_f32_16x16x4_f32(
      /*neg_a=*/false, a, /*neg_b=*/false, b,
      /*c_mod=*/(short)0, c, /*reuse_a=*/false, /*reuse_b=*/false);
}

__global__ __launch_bounds__(256)
void ssim_wmma_kernel(const float* __restrict__ X, const float* __restrict__ Y,
                      float* __restrict__ partial) {
  __shared__ float sm[SMEM_FLOATS];
  __shared__ float red[8];

  const int tid  = threadIdx.x;
  const int lane = tid & 31;
  const int w    = tid >> 5;
  const int m16  = lane & 15;   // M (or N) index within the 16x16 tile
  const int hi   = lane >> 4;   // 0: K pair {0,1} / rows 0..7 ; 1: {2,3} / 8..15

  const int bid    = blockIdx.x;
  const int img    = bid / (NSTRIPE * NSTRIP);
  const int rem    = bid % (NSTRIPE * NSTRIP);
  const int stripe = rem / NSTRIP;
  const int strip  = rem % NSTRIP;

  const int c0in = stripe * STRIPE_W;
  const size_t base = (size_t)img * (IMG * IMG);
  const float* xp = X + base;
  const float* yp = Y + base;

  // Band-weight fragments, shared by all WMMAs (vertical A / horizontal B).
  v2f bw[6];
#pragma unroll
  for (int q = 0; q < 6; ++q) {
    const int k0 = 4 * q + 2 * hi;
    bw[q].x = band7(m16, k0);
    bw[q].y = band7(m16, k0 + 1);
  }

  float accS = 0.0f;

  for (int mb = 0; mb < NB; ++mb) {
    const int r0 = strip * STRIP_H + mb * 16;           // top output row == top input row
    const int rows_valid = min(16, STRIP_H - mb * 16);  // last batch: 14

    // ---- step 1: stage 24 x 80 of x and y into LDS (rows>=22, cols>=70 -> 0)
    for (int idx = tid; idx < LROWS_PAD * LXW; idx += 256) {
      const int rr = idx / LXW;
      const int cc = idx - rr * LXW;
      float vx = 0.0f, vy = 0.0f;
      if (cc < INCOLS && rr < LROWS) {
        const int gr = min(r0 + rr, IMG - 1);
        const int gc = min(c0in + cc, IMG - 1);
        const size_t o = (size_t)gr * IMG + gc;
        vx = xp[o];
        vy = yp[o];
      }
      sm[LX_OFF + idx] = vx;
      sm[LY_OFF + idx] = vy;
    }
    __syncthreads();

    // ---- step 2: vertical 7-tap via WMMA -> Scol[5][16][SCW] ---------------
    // Scol[s][m][c] = sum_{k=m..m+6} stat_s(x[r0+k][c], y[r0+k][c])
    if (w < 5) {                                        // one col-block per wave
      const int cB = w * 16;
      v8f a0 = {0.f,0.f,0.f,0.f,0.f,0.f,0.f,0.f};
      v8f a1 = a0, a2 = a0, a3 = a0, a4 = a0;
#pragma unroll
      for (int q = 0; q < 6; ++q) {
        const int k0 = 4 * q + 2 * hi;
        const float x0 = sm[LX_OFF + k0 * LXW + cB + m16];
        const float y0 = sm[LY_OFF + k0 * LXW + cB + m16];
        const float x1 = sm[LX_OFF + (k0 + 1) * LXW + cB + m16];
        const float y1 = sm[LY_OFF + (k0 + 1) * LXW + cB + m16];
        v2f b;
        b.x = x0;      b.y = x1;      a0 = wmma4(bw[q], b, a0);
        b.x = y0;      b.y = y1;      a1 = wmma4(bw[q], b, a1);
        b.x = x0 * x0; b.y = x1 * x1; a2 = wmma4(bw[q], b, a2);
        b.x = y0 * y0; b.y = y1 * y1; a3 = wmma4(bw[q], b, a3);
        b.x = x0 * y0; b.y = x1 * y1; a4 = wmma4(bw[q], b, a4);
      }
      const int rb = hi * 8;
      const int cw = cB + m16;
#pragma unroll
      for (int j = 0; j < 8; ++j) {
        const int ro = (rb + j) * SCW + cw;
        sm[0 * (16 * SCW) + ro] = a0[j];
        sm[1 * (16 * SCW) + ro] = a1[j];
        sm[2 * (16 * SCW) + ro] = a2[j];
        sm[3 * (16 * SCW) + ro] = a3[j];
        sm[4 * (16 * SCW) + ro] = a4[j];
      }
    }
    __syncthreads();

    // ---- step 3: horizontal 7-tap via WMMA -> H[5][16][HW] -----------------
    // H[s][m][n] = sum_{j=0..6} Scol[s][m][c0+n+j]
    if (w < 4) {                                        // one out-block per wave
      const int c0 = w * 16;
      v8f a0 = {0.f,0.f,0.f,0.f,0.f,0.f,0.f,0.f};
      v8f a1 = a0, a2 = a0, a3 = a0, a4 = a0;
      const int rowb = m16 * SCW + c0;
#pragma unroll
      for (int q = 0; q < 6; ++q) {
        const int k0 = 4 * q + 2 * hi;
        v2f a;
        a.x = sm[0 * (16 * SCW) + rowb + k0];
        a.y = sm[0 * (16 * SCW) + rowb + k0 + 1];
        a0 = wmma4(a, bw[q], a0);
        a.x = sm[1 * (16 * SCW) + rowb + k0];
        a.y = sm[1 * (16 * SCW) + rowb + k0 + 1];
        a1 = wmma4(a, bw[q], a1);
        a.x = sm[2 * (16 * SCW) + rowb + k0];
        a.y = sm[2 * (16 * SCW) + rowb + k0 + 1];
        a2 = wmma4(a, bw[q], a2);
        a.x = sm[3 * (16 * SCW) + rowb + k0];
        a.y = sm[3 * (16 * SCW) + rowb + k0 + 1];
        a3 = wmma4(a, bw[q], a3);
        a.x = sm[4 * (16 * SCW) + rowb + k0];
        a.y = sm[4 * (16 * SCW) + rowb + k0 + 1];
        a4 = wmma4(a, bw[q], a4);
      }
      const int rb = hi * 8;
      const int cw = c0 + m16;
#pragma unroll
      for (int j = 0; j < 8; ++j) {
        const int ro = (rb + j) * HW + cw;
        sm[H_OFF + 0 * HSTAT + ro] = a0[j];
        sm[H_OFF + 1 * HSTAT + ro] = a1[j];
        sm[H_OFF + 2 * HSTAT + ro] = a2[j];
        sm[H_OFF + 3 * HSTAT + ro] = a3[j];
        sm[H_OFF + 4 * HSTAT + ro] = a4[j];
      }
    }
    __syncthreads();

    // ---- step 4: pointwise SSIM + accumulate -------------------------------
    {
      const float inv49 = 1.0f / 49.0f;
      const float covn  = 49.0f / 48.0f;
      const float C1 = 0.0001f, C2 = 0.0009f;
#pragma unroll
      for (int it = 0; it < 4; ++it) {
        const int pix = tid + it * 256;                 // 16*64 / 256 = 4 each
        const int mm = pix >> 6;
        const int cc = pix & 63;
        if (mm < rows_valid && (c0in + cc) < OUTD) {
          const int o = mm * HW + cc;
          const float ux  = sm[H_OFF + 0 * HSTAT + o] * inv49;
          const float uy  = sm[H_OFF + 1 * HSTAT + o] * inv49;
          const float uxx = sm[H_OFF + 2 * HSTAT + o] * inv49;
          const float uyy = sm[H_OFF + 3 * HSTAT + o] * inv49;
          const float uxy = sm[H_OFF + 4 * HSTAT + o] * inv49;
          const float vx  = covn * (uxx - ux * ux);
          const float vy  = covn * (uyy - uy * uy);
          const float vxy = covn * (uxy - ux * uy);
          const float num = (2.0f * ux * uy + C1) * (2.0f * vxy + C2);
          const float den = (ux * ux + uy * uy + C1) * (vx + vy + C2);
          accS += num / den;
        }
      }
    }
    __syncthreads();   // before next batch overwrites staging/H union
  }

  // ---- deterministic block reduction ---------------------------------------
  float v = accS;
#pragma unroll
  for (int o = 16; o > 0; o >>= 1) v += __shfl_down(v, o, 32);
  if (lane == 0) red[w] = v;
  __syncthreads();
  if (tid == 0) {
    float t = 0.0f;
#pragma unroll
    for (int i = 0; i < 8; ++i) t += red[i];
    partial[blockIdx.x] = t;
  }
}

__global__ __launch_bounds__(256)
void ssim_reduce_kernel(const float* __restrict__ partial, float* __restrict__ out,
                        int n) {
  __shared__ float s[256];
  float v = 0.0f;
  for (int i = threadIdx.x; i < n; i += 256) v += partial[i];
  s[threadIdx.x] = v;
  __syncthreads();
  for (int o = 128; o > 0; o >>= 1) {
    if (threadIdx.x < o) s[threadIdx.x] += s[threadIdx.x + o];
    __syncthreads();
  }
  if (threadIdx.x == 0)
    out[0] = s[0] * (1.0f / ((float)OUTD * (float)OUTD * (float)NIMG));
}

extern "C" void kernel_launch(void* const* d_in, const int* in_sizes, int n_in,
                              void* d_out, int out_size, void* d_ws, size_t ws_size,
                              hipStream_t stream) {
  const float* pred   = (const float*)d_in[0];
  const float* actual = (const float*)d_in[1];
  float* out = (float*)d_out;
  float* ws  = (float*)d_ws;                 // needs NBLK*4 = 9216 bytes

  ssim_wmma_kernel<<<dim3(NBLK), dim3(256), 0, stream>>>(pred, actual, ws);
  ssim_reduce_kernel<<<dim3(1), dim3(256), 0, stream>>>(ws, out, NBLK);
}